// KDPointToPointLoss_26371099197709
// MI455X (gfx1250) — compile-verified
//
#include <hip/hip_runtime.h>

// CDNA5 / gfx1250, wave32. Brute-force NN point-cloud loss.
//   score[s][t] = ||t||^2 - 2 s.t  via v_wmma_f32_16x16x4_f32:
//     A row = [sx, sy, sz, 1]
//     B col = [-2tx, -2ty, -2tz, valid_t ? ||t||^2 : +INF]   (precomputed pack)
//   Track only min_t score per source:  ||s - t*||^2 = ||s||^2 + min_t score.
// Inline-asm v_min3_num_f32 avoids LLVM's minnum canonicalization (v_max v,v,v spam).

typedef __attribute__((ext_vector_type(2))) float v2f;
typedef __attribute__((ext_vector_type(4))) float v4f;
typedef __attribute__((ext_vector_type(8))) float v8f;

#define BATCH 4
#define NPTS  8192
#define MPTS  8192
#define TILE  16
#define ST    2              // source tiles (16 rows each) per wave share each B operand
#define WAVES_PER_BLOCK 8

// raw CDNA5 min ops: no NaN canonicalization (inputs are never sNaN here)
__device__ __forceinline__ float min3f(float x, float y, float z) {
  float r;
  asm("v_min3_num_f32 %0, %1, %2, %3" : "=v"(r) : "v"(x), "v"(y), "v"(z));
  return r;
}
__device__ __forceinline__ float min2f(float x, float y) {
  float r;
  asm("v_min_num_f32 %0, %1, %2" : "=v"(r) : "v"(x), "v"(y));
  return r;
}

__global__ void kdl_init_ws(float* __restrict__ acc) {
  int i = threadIdx.x;
  if (i < 2 * BATCH) acc[i] = 0.0f;
}

// Pack each target point into WMMA-B form: pack[2m+0] = (-2tx, -2ty),
// pack[2m+1] = (-2tz, valid ? ||t||^2 : +INF). One float4 store per target.
__global__ __launch_bounds__(256) void kdl_pack_tgt(const float* __restrict__ tgt,
                                                    v4f* __restrict__ pack) {
  const int i = blockIdx.x * blockDim.x + threadIdx.x;   // flat b*M + m
  const float* tp = tgt + (size_t)i * 3;
  const float tx = tp[0], ty = tp[1], tz = tp[2];
  const float tsq = tx * tx + ty * ty + tz * tz;
  const bool validT = (tx != 0.0f) || (ty != 0.0f) || (tz != 0.0f);
  v4f p;
  p[0] = -2.0f * tx;
  p[1] = -2.0f * ty;
  p[2] = -2.0f * tz;
  p[3] = validT ? tsq : __builtin_inff();
  pack[i] = p;
}

__global__ __launch_bounds__(256) void kdl_nn_kernel(const float* __restrict__ src,
                                                     const v2f* __restrict__ pack,
                                                     float* __restrict__ acc) {
  const int lane = threadIdx.x & 31;
  const int wave = threadIdx.x >> 5;
  const int wtile = blockIdx.x * WAVES_PER_BLOCK + wave;  // one (ST*16)-row block per wave
  const int wtilesPerBatch = NPTS / (ST * TILE);
  const int b    = wtile / wtilesPerBatch;
  const int srow = (wtile % wtilesPerBatch) * (ST * TILE);

  const int col  = lane & 15;   // column index within 16-wide tile
  const int half = lane >> 4;   // K-split half for A/B operands (wave32 layout)

  const float* __restrict__ sb = src + (size_t)b * NPTS * 3;
  // per-lane B stream: entry 2m+half, zero VALU prep in the loop
  const v2f* __restrict__ pb = pack + ((size_t)b * MPTS * 2) + half;

  // ---- A operands (constant across the M loop), one per source tile ----
  // lane L holds source row M = L%16; lanes 0-15 supply K=0,1 ; lanes 16-31 K=2,3
  v2f a[ST];
#pragma unroll
  for (int st = 0; st < ST; ++st) {
    const float* sp = sb + (size_t)(srow + st * TILE + col) * 3;
    const float sx = sp[0], sy = sp[1], sz = sp[2];
    a[st][0] = half ? sz   : sx;   // K=2 : K=0
    a[st][1] = half ? 1.0f : sy;   // K=3 : K=1
  }

  // running min score for 8 rows per source tile (targets == col mod 16)
  float bestv[ST][8];
#pragma unroll
  for (int st = 0; st < ST; ++st)
#pragma unroll
    for (int k = 0; k < 8; ++k) bestv[st][k] = __builtin_inff();

  for (int tbase = 0; tbase < MPTS; tbase += 2 * TILE) {
    const v2f bm0 = pb[2 * (size_t)(tbase + col)];
    const v2f bm1 = pb[2 * (size_t)(tbase + TILE + col)];

#pragma unroll
    for (int st = 0; st < ST; ++st) {
      v8f c0 = {}, c1 = {};
      c0 = __builtin_amdgcn_wmma_f32_16x16x4_f32(false, a[st], false, bm0,
                                                 (short)0, c0, false, false);
      c1 = __builtin_amdgcn_wmma_f32_16x16x4_f32(false, a[st], false, bm1,
                                                 (short)0, c1, false, false);
#pragma unroll
      for (int k = 0; k < 8; ++k)
        bestv[st][k] = min3f(bestv[st][k], c0[k], c1[k]);
    }
  }

  // ---- butterfly min across the 16 lanes of each half-wave ----
  // xor masks 1,2,4,8 stay inside lanes [0..15] and [16..31] on wave32
#pragma unroll
  for (int off = 1; off < 16; off <<= 1) {
#pragma unroll
    for (int st = 0; st < ST; ++st)
#pragma unroll
      for (int k = 0; k < 8; ++k)
        bestv[st][k] = min2f(bestv[st][k], __shfl_xor(bestv[st][k], off, 32));
  }

  // lanes 0 (rows 0-7 of each tile) and 16 (rows 8-15): finalize + accumulate
  if (col == 0) {
    float lsum = 0.0f, lcnt = 0.0f;
#pragma unroll
    for (int st = 0; st < ST; ++st) {
#pragma unroll
      for (int k = 0; k < 8; ++k) {
        const int row = srow + st * TILE + half * 8 + k;
        const float* s2 = sb + (size_t)row * 3;
        const float ax = s2[0], ay = s2[1], az = s2[2];
        const bool validS = (ax != 0.0f) || (ay != 0.0f) || (az != 0.0f);
        const float ssq = ax * ax + ay * ay + az * az;
        float sq = ssq + bestv[st][k];          // ||s - t*||^2
        sq = sq > 0.0f ? sq : 0.0f;             // clamp cancellation residue
        if (validS) { lsum += sq; lcnt += 1.0f; }
      }
    }
    atomicAdd(&acc[b], lsum);
    atomicAdd(&acc[BATCH + b], lcnt);
  }
}

__global__ void kdl_finish(const float* __restrict__ acc, float* __restrict__ out) {
  if (threadIdx.x == 0 && blockIdx.x == 0) {
    float s = 0.0f;
    for (int b = 0; b < BATCH; ++b) {
      float cnt = acc[BATCH + b];
      cnt = cnt > 1.0f ? cnt : 1.0f;
      s += acc[b] / (3.0f * cnt);
    }
    out[0] = s / (float)BATCH;
  }
}

extern "C" void kernel_launch(void* const* d_in, const int* in_sizes, int n_in,
                              void* d_out, int out_size, void* d_ws, size_t ws_size,
                              hipStream_t stream) {
  (void)in_sizes; (void)n_in; (void)out_size; (void)ws_size;
  const float* src = (const float*)d_in[0];  // [B, N, 3] f32
  const float* tgt = (const float*)d_in[1];  // [B, M, 3] f32
  float* out = (float*)d_out;                // scalar f32

  // ws layout: [0 .. B*M*16) packed B operands, then 2*B floats of accumulators
  v4f*   pack = (v4f*)d_ws;
  float* acc  = (float*)((char*)d_ws + (size_t)BATCH * MPTS * sizeof(v4f));

  kdl_init_ws<<<1, 32, 0, stream>>>(acc);
  kdl_pack_tgt<<<(BATCH * MPTS) / 256, 256, 0, stream>>>(tgt, pack);

  const int waves  = BATCH * (NPTS / (ST * TILE));   // 1024 waves, 32 rows each
  const int blocks = waves / WAVES_PER_BLOCK;        // 128 blocks x 256 threads
  kdl_nn_kernel<<<blocks, WAVES_PER_BLOCK * 32, 0, stream>>>(src, (const v2f*)pack, acc);

  kdl_finish<<<1, 1, 0, stream>>>(acc, out);
}